// ModelStep_12120397709837
// MI455X (gfx1250) — compile-verified
//
#include <hip/hip_runtime.h>
#include <math.h>

#define BLOCK 256
#define NS 6
#define MXSTEP_CAP 5000000
#define ATOL_F 1e-6f
#define RTOL_F 1e-6f

// ---------------- gfx1250 async global<->LDS helpers ----------------
typedef __attribute__((address_space(3))) void lds_void_t;

__device__ __forceinline__ unsigned to_lds_addr(void* p) {
  // generic -> LDS addrspace cast; numeric value of an AS(3) pointer is the LDS byte offset
  return (unsigned)(unsigned long long)(lds_void_t*)p;
}

__device__ __forceinline__ void async_load_b32(unsigned lds_addr, const void* gptr) {
  unsigned long long ga = (unsigned long long)gptr;
  asm volatile("global_load_async_to_lds_b32 %0, %1, off"
               :: "v"(lds_addr), "v"(ga) : "memory");
}

__device__ __forceinline__ void async_store_b32(const void* gptr, unsigned lds_addr) {
  unsigned long long ga = (unsigned long long)gptr;
  asm volatile("global_store_async_from_lds_b32 %0, %1, off"
               :: "v"(ga), "v"(lds_addr) : "memory");
}

__device__ __forceinline__ void wait_async0() {
  asm volatile("s_wait_asynccnt 0" ::: "memory");
}

// fast hw transcendentals (v_log_f32 / v_exp_f32): x>0 guaranteed at call sites
__device__ __forceinline__ float fast_pow_pos(float x, float p) {
  return __builtin_amdgcn_exp2f(p * __builtin_amdgcn_logf(x));
}

// ---------------- folded rate coefficients ----------------
struct RateK {
  float A0, P0, P3;        // dCp  = A0 + P0*y0 + P3*y3
  float B0, B1;            // dCbc = B0 + B1*y1
  float E0, E2, E3, E4;    // dCeu = E0*y0 + E2*y2 + E3*y3 + E4*y4 - q
  float KQ;                // q = KQ * y2 * y5
  float G3;                // dCeb = q - G3*y3 ; dFc = -dCeb
  float I0, I3, I4;        // dCis = I0*y0 + I3*y3 + I4*y4
};

__device__ __forceinline__ RateK make_coeffs(const float* c) {
  float c0 = c[0],  c1 = c[1],  c6 = c[6],  c7 = c[7],  c8 = c[8];
  float c9 = c[9],  c10 = c[10], c11 = c[11], c12 = c[12], c13 = c[13];
  float c14 = c[14], c15 = c[15], c24 = c[24], c25 = c[25], c26 = c[26], c27 = c[27];
  RateK K;
  K.A0 = c6 * c0 / c24;
  K.P0 = -(((c6 - c8) + (1.f - c14) * c8 + c12) / (1000.f * c24));
  K.P3 = c12 * c13 / (5.f * c24);
  K.B0 = c7 * c1 / c25;
  K.B1 = -c7 / (55.f * c25);
  K.E0 = c12 / (1000.f * c27);
  K.E2 = -c11 / 5.f;
  K.E3 = c10 / 5.f;
  K.E4 = c12 / (45.f * c27);
  K.KQ = c9 / 25.f;
  K.G3 = c10 / 5.f + c12 / (5.f * c27);
  K.I0 = (1.f - c14) * c8 / (1000.f * c26);
  K.I3 = c12 * (1.f - c13) / (5.f * c26);
  K.I4 = -(((1.f - c15) * c8 + c12) / (45.f * c26));
  return K;
}

__device__ __forceinline__ void rate6(const RateK& K, const float y[NS], float d[NS]) {
  float q = K.KQ * y[2] * y[5];
  d[0] = K.A0 + K.P0 * y[0] + K.P3 * y[3];
  d[1] = K.B0 + K.B1 * y[1];
  d[2] = K.E0 * y[0] + K.E2 * y[2] + K.E3 * y[3] + K.E4 * y[4] - q;
  float g = q - K.G3 * y[3];
  d[3] = g;
  d[4] = K.I0 * y[0] + K.I3 * y[3] + K.I4 * y[4];
  d[5] = -g;
}

// ---------------- setup: fold coefficients once into workspace ----------------
__global__ void setup_kernel(const float* __restrict__ c, const float* __restrict__ t_in,
                             const float* __restrict__ dT_in, float* __restrict__ kc) {
  if (threadIdx.x == 0) {
    RateK K = make_coeffs(c);
    kc[0] = K.A0; kc[1] = K.P0; kc[2] = K.P3; kc[3] = K.B0; kc[4] = K.B1;
    kc[5] = K.E0; kc[6] = K.E2; kc[7] = K.E3; kc[8] = K.E4; kc[9] = K.KQ;
    kc[10] = K.G3; kc[11] = K.I0; kc[12] = K.I3; kc[13] = K.I4;
    kc[14] = t_in[0];
    kc[15] = t_in[0] + dT_in[0];
  }
}

// ---------------- main dopri5 kernel: one thread per ODE element ----------------
template <bool FROMWS>
__global__ __launch_bounds__(BLOCK) void dopri5_kernel(
    const float* __restrict__ y_in, const float* __restrict__ c,
    const float* __restrict__ t_in, const float* __restrict__ dT_in,
    const float* __restrict__ kc, float* __restrict__ y_out, int B)
{
  __shared__ float smem[BLOCK * NS];
  const int tid = threadIdx.x;
  const long long base  = (long long)blockIdx.x * (BLOCK * NS);
  const long long total = (long long)B * NS;
  const long long elem  = (long long)blockIdx.x * BLOCK + tid;
  const bool full = (base + (long long)(BLOCK * NS)) <= total;  // uniform per block

  // ---- async coalesced stage-in: global -> LDS (gfx1250 ASYNCcnt path) ----
  if (full) {
    #pragma unroll
    for (int j = 0; j < NS; ++j)
      async_load_b32(to_lds_addr(&smem[j * BLOCK + tid]), y_in + base + j * BLOCK + tid);
  } else {
    #pragma unroll
    for (int j = 0; j < NS; ++j) {
      long long idx = base + j * BLOCK + tid;
      if (idx < total)
        async_load_b32(to_lds_addr(&smem[j * BLOCK + tid]), y_in + idx);
    }
  }

  // overlap uniform coefficient loads with the in-flight async LDS fill
  RateK K;
  float t0, target;
  if (FROMWS) {
    K.A0 = kc[0]; K.P0 = kc[1]; K.P3 = kc[2]; K.B0 = kc[3]; K.B1 = kc[4];
    K.E0 = kc[5]; K.E2 = kc[6]; K.E3 = kc[7]; K.E4 = kc[8]; K.KQ = kc[9];
    K.G3 = kc[10]; K.I0 = kc[11]; K.I3 = kc[12]; K.I4 = kc[13];
    t0 = kc[14]; target = kc[15];
  } else {
    K = make_coeffs(c);
    t0 = t_in[0];
    target = t0 + dT_in[0];
  }

  wait_async0();
  __syncthreads();

  if (elem < B) {
    float y[NS];
    #pragma unroll
    for (int s = 0; s < NS; ++s) y[s] = smem[tid * NS + s];

    float f[NS];
    rate6(K, y, f);

    // ---- Hairer initial step size (order=4 as in jax odeint) ----
    float dt;
    {
      float scale[NS], d0s = 0.f, d1s = 0.f;
      #pragma unroll
      for (int s = 0; s < NS; ++s) {
        scale[s] = ATOL_F + fabsf(y[s]) * RTOL_F;
        float a = y[s] / scale[s]; d0s += a * a;
        float b = f[s] / scale[s]; d1s += b * b;
      }
      float d0 = sqrtf(d0s), d1 = sqrtf(d1s);
      float h0 = ((d0 < 1e-5f) || (d1 < 1e-5f)) ? 1e-6f : 0.01f * d0 / d1;
      float yt[NS], f1[NS];
      #pragma unroll
      for (int s = 0; s < NS; ++s) yt[s] = y[s] + h0 * f[s];
      rate6(K, yt, f1);
      float d2s = 0.f;
      #pragma unroll
      for (int s = 0; s < NS; ++s) { float a = (f1[s] - f[s]) / scale[s]; d2s += a * a; }
      float d2 = sqrtf(d2s) / h0;
      float h1 = ((d1 <= 1e-15f) && (d2 <= 1e-15f))
                     ? fmaxf(1e-6f, h0 * 1e-3f)
                     : fast_pow_pos(0.01f / fmaxf(d1, d2), 0.2f);  // 1/(order+1), order=4
      dt = fminf(100.f * h0, h1);
      dt = fmaxf(dt, 0.f);
    }

    // ---- Dormand-Prince tableau ----
    const float B21 = 0.2f;
    const float B31 = 3.f/40.f,  B32 = 9.f/40.f;
    const float B41 = 44.f/45.f, B42 = -56.f/15.f, B43 = 32.f/9.f;
    const float B51 = 19372.f/6561.f, B52 = -25360.f/2187.f, B53 = 64448.f/6561.f, B54 = -212.f/729.f;
    const float B61 = 9017.f/3168.f,  B62 = -355.f/33.f,  B63 = 46732.f/5247.f,
                B64 = 49.f/176.f,     B65 = -5103.f/18656.f;
    const float C1s = 35.f/384.f, C3s = 500.f/1113.f, C4s = 125.f/192.f,
                C5s = -2187.f/6784.f, C6s = 11.f/84.f;
    const float E1c = (float)(35.0/384.0 - 1951.0/21600.0);
    const float E3c = (float)(500.0/1113.0 - 22642.0/50085.0);
    const float E4c = (float)(125.0/192.0 - 451.0/720.0);
    const float E5c = (float)(-2187.0/6784.0 + 12231.0/42400.0);
    const float E6c = (float)(11.0/84.0 - 649.0/6300.0);
    const float E7c = (float)(-1.0/60.0);
    const float M1 = (float)(6025192743.0/30085553152.0/2.0);
    const float M3 = (float)(51252292925.0/65400821598.0/2.0);
    const float M4 = (float)(-2691868925.0/45128329728.0/2.0);
    const float M5 = (float)(187940372067.0/1594534317056.0/2.0);
    const float M6 = (float)(-1776094331.0/19743644256.0/2.0);
    const float M7 = (float)(11237099.0/235043384.0/2.0);

    float t = t0;
    float yout[NS];
    bool have = false;
    int it = 0;

    while ((t < target) && (it < MXSTEP_CAP) && (dt > 0.f)) {
      float k2[NS], k3[NS], k4[NS], k5[NS], k6[NS], k7[NS], yn[NS], yt[NS];

      #pragma unroll
      for (int s = 0; s < NS; ++s) yt[s] = y[s] + dt * (B21 * f[s]);
      rate6(K, yt, k2);
      #pragma unroll
      for (int s = 0; s < NS; ++s) yt[s] = y[s] + dt * (B31*f[s] + B32*k2[s]);
      rate6(K, yt, k3);
      #pragma unroll
      for (int s = 0; s < NS; ++s) yt[s] = y[s] + dt * (B41*f[s] + B42*k2[s] + B43*k3[s]);
      rate6(K, yt, k4);
      #pragma unroll
      for (int s = 0; s < NS; ++s) yt[s] = y[s] + dt * (B51*f[s] + B52*k2[s] + B53*k3[s] + B54*k4[s]);
      rate6(K, yt, k5);
      #pragma unroll
      for (int s = 0; s < NS; ++s) yt[s] = y[s] + dt * (B61*f[s] + B62*k2[s] + B63*k3[s] + B64*k4[s] + B65*k5[s]);
      rate6(K, yt, k6);
      #pragma unroll
      for (int s = 0; s < NS; ++s) yn[s] = y[s] + dt * (C1s*f[s] + C3s*k3[s] + C4s*k4[s] + C5s*k5[s] + C6s*k6[s]);
      rate6(K, yn, k7);  // FSAL stage

      // error ratio^2 (jax: ratio = sqrt(mean((err/tol)^2)); compare via errsum<=6)
      float errsum = 0.f;
      #pragma unroll
      for (int s = 0; s < NS; ++s) {
        float err = dt * (E1c*f[s] + E3c*k3[s] + E4c*k4[s] + E5c*k5[s] + E6c*k6[s] + E7c*k7[s]);
        float tol = ATOL_F + RTOL_F * fmaxf(fabsf(y[s]), fabsf(yn[s]));
        float rr = err / tol;
        errsum += rr * rr;
      }
      float ratio_sq = errsum * (1.f / 6.f);
      bool accept = (errsum <= 6.f);          // ratio <= 1
      float next_t = t + dt;

      // jax optimal_step_size: safety=0.9, ifactor=10, dfactor=0.2, order=5
      // ratio^(-1/5) = exp2(-0.1 * log2(ratio^2))
      float dfac = (ratio_sq < 1.f) ? 1.f : 0.2f;
      float fac  = fminf(10.f, fmaxf(0.9f * __builtin_amdgcn_exp2f(
                                       -0.1f * __builtin_amdgcn_logf(ratio_sq)), dfac));
      float ndt  = (errsum == 0.f) ? dt * 10.f : dt * fac;
      ndt = fmaxf(ndt, 0.f);

      if (accept) {
        if (!(next_t < target)) {
          // final accepted step: x is already known -> evaluate jax's 4th-order
          // interpolation polynomial right here, carry only yout[6]
          float x = (target - t) / dt;
          #pragma unroll
          for (int s = 0; s < NS; ++s) {
            float ymid = y[s] + dt * (M1*f[s] + M3*k3[s] + M4*k4[s] + M5*k5[s] + M6*k6[s] + M7*k7[s]);
            float dy0 = dt * f[s];
            float dy1 = dt * k7[s];
            float pa = -2.f*dy0 + 2.f*dy1 - 8.f*y[s] - 8.f*yn[s] + 16.f*ymid;
            float pb =  5.f*dy0 - 3.f*dy1 + 18.f*y[s] + 14.f*yn[s] - 32.f*ymid;
            float pc = -4.f*dy0 +     dy1 - 11.f*y[s] -  5.f*yn[s] + 16.f*ymid;
            yout[s] = ((((pa*x + pb)*x + pc)*x + dy0)*x + y[s]);
          }
          have = true;
        }
        #pragma unroll
        for (int s = 0; s < NS; ++s) { y[s] = yn[s]; f[s] = k7[s]; }
        t = next_t;
      }
      dt = ndt;
      ++it;
    }

    if (!have) {
      #pragma unroll
      for (int s = 0; s < NS; ++s) yout[s] = y[s];
    }
    #pragma unroll
    for (int s = 0; s < NS; ++s) smem[tid * NS + s] = yout[s];
  }

  // ---- async coalesced stage-out: LDS -> global ----
  __syncthreads();
  if (full) {
    #pragma unroll
    for (int j = 0; j < NS; ++j)
      async_store_b32(y_out + base + j * BLOCK + tid, to_lds_addr(&smem[j * BLOCK + tid]));
  } else {
    #pragma unroll
    for (int j = 0; j < NS; ++j) {
      long long idx = base + j * BLOCK + tid;
      if (idx < total)
        async_store_b32(y_out + idx, to_lds_addr(&smem[j * BLOCK + tid]));
    }
  }
  wait_async0();
}

// ---------------- tuple tail: (w_new empty), c passthrough, t_new ----------------
__global__ void tail_kernel(const float* __restrict__ c,
                            const float* __restrict__ t_in,
                            const float* __restrict__ dT_in,
                            float* __restrict__ out_tail)
{
  int i = threadIdx.x;
  if (i < 28) out_tail[i] = c[i];
  if (i == 28) out_tail[28] = t_in[0] + dT_in[0];
}

extern "C" void kernel_launch(void* const* d_in, const int* in_sizes, int n_in,
                              void* d_out, int out_size, void* d_ws, size_t ws_size,
                              hipStream_t stream) {
  (void)n_in; (void)out_size;
  const float* y  = (const float*)d_in[0];
  // d_in[1] = w, zero elements
  const float* c  = (const float*)d_in[2];
  const float* t  = (const float*)d_in[3];
  const float* dT = (const float*)d_in[4];
  const int B = in_sizes[0] / NS;
  float* out = (float*)d_out;

  const int grid = (B + BLOCK - 1) / BLOCK;
  if (ws_size >= 16 * sizeof(float) && d_ws != nullptr) {
    float* kc = (float*)d_ws;
    setup_kernel<<<1, 1, 0, stream>>>(c, t, dT, kc);
    dopri5_kernel<true><<<grid, BLOCK, 0, stream>>>(y, c, t, dT, kc, out, B);
  } else {
    dopri5_kernel<false><<<grid, BLOCK, 0, stream>>>(y, c, t, dT, nullptr, out, B);
  }
  tail_kernel<<<1, 32, 0, stream>>>(c, t, dT, out + (size_t)B * NS);
}